// WCEL_Loss_24893630447739
// MI455X (gfx1250) — compile-verified
//
#include <hip/hip_runtime.h>

#define Bb 4
#define Cc 150
#define Hh 384
#define Ww 512
#define HW (Hh * Ww)
#define NPIX (Bb * HW)

typedef __attribute__((ext_vector_type(2))) float v2f;
typedef __attribute__((ext_vector_type(8))) float v8f;

// ---------------- init: zero the accumulators in workspace ----------------
__global__ void wcel_init(double* lsum, unsigned int* cnt) {
    if (threadIdx.x == 0) { *lsum = 0.0; *cnt = 0u; }
}

// ---------------- rowsum[g] = sum_c W[g,c] via V_WMMA_F32_16X16X4_F32 -----
// One block of 320 threads = 10 waves; wave w computes rows [16w, 16w+16).
// A-matrix (16x4 f32): lanes 0-15 hold (M=lane, K=0/1) in vgpr0/1,
//                      lanes 16-31 hold (M=lane-16, K=2/3).
// B = ones(4x16)  =>  D[m,n] = rowsum of the 4-wide K chunk, accumulated in C.
__global__ __launch_bounds__(320) void wcel_rowsum_wmma(
    const float* __restrict__ w, float* __restrict__ rowsum) {
    const int lane = threadIdx.x & 31;
    const int wave = threadIdx.x >> 5;
    const int row0 = wave * 16;
    const int half = (lane >> 4) & 1;        // 0: K=0/1, 1: K=2/3
    const int mrow = row0 + (lane & 15);

    v8f acc = {};
    v2f bones; bones.x = 1.0f; bones.y = 1.0f;

    for (int k0 = 0; k0 < Cc; k0 += 4) {
        const int kk = k0 + (half ? 2 : 0);
        v2f a;
        a.x = (mrow < Cc && kk     < Cc) ? w[mrow * Cc + kk    ] : 0.0f;
        a.y = (mrow < Cc && kk + 1 < Cc) ? w[mrow * Cc + kk + 1] : 0.0f;
        acc = __builtin_amdgcn_wmma_f32_16x16x4_f32(
            /*neg_a=*/false, a, /*neg_b=*/false, bones,
            /*c_mod=*/(short)0, acc, /*reuse_a=*/false, /*reuse_b=*/false);
    }

    // D layout (16x16 f32): lane 0 holds (M=r, N=0) in vgpr r (r=0..7),
    //                       lane 16 holds (M=r+8, N=0) in vgpr r.
    if ((lane & 15) == 0) {
        const int rbase = row0 + (half ? 8 : 0);
#pragma unroll
        for (int r = 0; r < 8; ++r) {
            const int row = rbase + r;
            if (row < Cc) rowsum[row] = acc[r];
        }
    }
}

// ---------------- main streaming pass: online logsumexp + weighted dot ----
__global__ __launch_bounds__(256) void wcel_main(
    const float* __restrict__ logit, const int* __restrict__ gtb,
    const float* __restrict__ mask, const float* __restrict__ w,
    const float* __restrict__ rowsum,
    double* __restrict__ lsum, unsigned int* __restrict__ cnt) {
    // Stage the full 150x150 weight matrix in LDS (90 KB of 320 KB/WGP).
    __shared__ float wlds[Cc * Cc];
    for (int i = threadIdx.x; i < Cc * Cc; i += 256) wlds[i] = w[i];
    __syncthreads();

    const int pix = blockIdx.x * 256 + threadIdx.x;   // (b,h,w) flat order
    const int b   = pix / HW;
    const int hw  = pix - b * HW;
    const size_t base = (size_t)b * Cc * HW + (size_t)hw;

    const int   g = gtb[pix];
    const float m = mask[pix];
    const float* __restrict__ wrow = &wlds[g * Cc];

    float runmax = -__builtin_inff();
    float s = 0.0f, dot = 0.0f;

    for (int c = 0; c < Cc; ++c) {
        const float x = logit[base + (size_t)c * HW];
        if (c + 4 < Cc)
            __builtin_prefetch(&logit[base + (size_t)(c + 4) * HW], 0, 1);
        dot = fmaf(wrow[c], x, dot);
        const float nm = fmaxf(runmax, x);
        s = s * __expf(runmax - nm) + __expf(x - nm);
        runmax = nm;
    }

    const float lse = runmax + __logf(s);
    float        pl = m * (lse * rowsum[g] - dot);   // = -sum_c w*lp for this pixel
    unsigned int v  = (m > 0.0f) ? 1u : 0u;

    // wave32 reduction
#pragma unroll
    for (int off = 16; off > 0; off >>= 1) {
        pl += __shfl_down(pl, off, 32);
        v  += __shfl_down(v,  off, 32);
    }

    __shared__ double       bsum[8];
    __shared__ unsigned int bcnt[8];
    const int lane = threadIdx.x & 31, wave = threadIdx.x >> 5;
    if (lane == 0) { bsum[wave] = (double)pl; bcnt[wave] = v; }
    __syncthreads();
    if (threadIdx.x == 0) {
        double t = 0.0; unsigned int tv = 0u;
#pragma unroll
        for (int i = 0; i < 8; ++i) { t += bsum[i]; tv += bcnt[i]; }
        atomicAdd(lsum, t);          // global_atomic_add_f64
        atomicAdd(cnt, tv);
    }
}

// ---------------- finalize: loss = sum / valid ----------------------------
__global__ void wcel_finalize(const double* lsum, const unsigned int* cnt,
                              float* out) {
    if (threadIdx.x == 0) {
        const unsigned int c = *cnt;
        out[0] = (float)(*lsum / (double)(c ? c : 1u));
    }
}

extern "C" void kernel_launch(void* const* d_in, const int* in_sizes, int n_in,
                              void* d_out, int out_size, void* d_ws, size_t ws_size,
                              hipStream_t stream) {
    const float* logit = (const float*)d_in[0];   // [B,C,H,W] f32
    const int*   gtb   = (const int*)d_in[1];     // [B,1,H,W] i32
    // d_in[2] = gt (unused by the loss math)
    const float* mask  = (const float*)d_in[3];   // [B,1,H,W] f32
    const float* w     = (const float*)d_in[4];   // [C,C] f32 (row-normalized)

    char*         ws     = (char*)d_ws;
    double*       lsum   = (double*)ws;                 // offset 0
    unsigned int* cnt    = (unsigned int*)(ws + 8);     // offset 8
    float*        rowsum = (float*)(ws + 16);           // offset 16, 150 floats

    wcel_init<<<1, 64, 0, stream>>>(lsum, cnt);
    wcel_rowsum_wmma<<<1, 320, 0, stream>>>(w, rowsum);
    wcel_main<<<NPIX / 256, 256, 0, stream>>>(logit, gtb, mask, w, rowsum,
                                              lsum, cnt);
    wcel_finalize<<<1, 32, 0, stream>>>(lsum, cnt, (float*)d_out);
}